// IGNet_38869454028881
// MI455X (gfx1250) — compile-verified
//
#include <hip/hip_runtime.h>
#include <hip/hip_bf16.h>
#include <math.h>

#define NV 300
#define NA 12
#define ND 4
#define VAE (NV*NA)      // 3600
#define NS 1024
#define NP 1500
#define NPPAD 1536       // 96 tiles of 16
#define NTILE 96
#define OUTW 3620        // 4 + 4 + 3600 + 9 + 3
#define PI_F 3.14159265358979323846f

typedef __attribute__((ext_vector_type(2))) float v2f;
typedef __attribute__((ext_vector_type(8))) float v8f;

// ---------------------------------------------------------------- constants
__device__ __forceinline__ void fib_view(int i, float& x, float& y, float& z) {
    const float phi = 0.61803398874989484820f;          // (sqrt(5)-1)/2
    float zi = (2.0f * (float)i + 1.0f) / (float)NV - 1.0f;
    float r  = sqrtf(fmaxf(1.0f - zi * zi, 0.0f));
    float a  = 2.0f * PI_F * (float)i * phi;
    x = r * cosf(a);
    y = r * sinf(a);
    z = zi;
}

// views (300x3), GR (3600x3x3 row-major), pk (3600x12 canonical key points)
__global__ void k_setup(float* views, float* GRm, float* pkbuf) {
    int n = blockIdx.x * blockDim.x + threadIdx.x;
    if (n >= VAE) return;
    if (n < NV) {
        float x, y, z; fib_view(n, x, y, z);
        views[n*3+0] = x; views[n*3+1] = y; views[n*3+2] = z;
    }
    int vi = n / NA, ai = n % NA;
    float vx, vy, vz; fib_view(vi, vx, vy, vz);
    // towards = -view
    float ax0 = -vx, ax1 = -vy, ax2 = -vz;
    float ay0 = -ax1, ay1 = ax0, ay2 = 0.0f;
    float ayn = sqrtf(ay0*ay0 + ay1*ay1 + ay2*ay2);
    if (ayn < 1e-6f) { ay0 = 0.0f; ay1 = 1.0f; ay2 = 0.0f; }
    float axn = sqrtf(ax0*ax0 + ax1*ax1 + ax2*ax2);
    ax0 /= axn; ax1 /= axn; ax2 /= axn;
    ayn = sqrtf(ay0*ay0 + ay1*ay1 + ay2*ay2);
    ay0 /= ayn; ay1 /= ayn; ay2 /= ayn;
    float az0 = ax1*ay2 - ax2*ay1;
    float az1 = ax2*ay0 - ax0*ay2;
    float az2 = ax0*ay1 - ax1*ay0;
    float ang = (PI_F / (float)NA) * (float)ai;
    float s = sinf(ang), c = cosf(ang);
    // G = R2 @ R1 ; columns: ax, c*ay+s*az, -s*ay+c*az
    float G[9];
    G[0] = ax0; G[1] = c*ay0 + s*az0; G[2] = -s*ay0 + c*az0;
    G[3] = ax1; G[4] = c*ay1 + s*az1; G[5] = -s*ay1 + c*az1;
    G[6] = ax2; G[7] = c*ay2 + s*az2; G[8] = -s*ay2 + c*az2;
    float* gd = GRm + (size_t)n * 9;
    #pragma unroll
    for (int k = 0; k < 9; ++k) gd[k] = G[k];
    // canonical key points: local = {(.02,0,0),(.02,.01,0),(.02,-.01,0),(0,0,0)}
    const float lx[4] = {0.02f, 0.02f, 0.02f, 0.0f};
    const float ly[4] = {0.0f,  0.01f, -0.01f, 0.0f};
    float* pk = pkbuf + (size_t)n * 12;
    #pragma unroll
    for (int k = 0; k < 4; ++k)
        #pragma unroll
        for (int i = 0; i < 3; ++i)
            pk[k*3 + i] = G[i*3+0]*lx[k] + G[i*3+1]*ly[k];
}

// ---------------------------------------------------------------- view knn
__global__ void k_views_trans(const float* pose, const float* views, float* vt, int B) {
    int id = blockIdx.x * blockDim.x + threadIdx.x;
    if (id >= B * NV) return;
    int b = id / NV, v = id % NV;
    const float* P = pose + (size_t)b * 12;  // 3x4 row-major
    const float* w = views + v*3;
    #pragma unroll
    for (int i = 0; i < 3; ++i)
        vt[(size_t)id*3 + i] = w[0]*P[i*4+0] + w[1]*P[i*4+1] + w[2]*P[i*4+2];
}

__global__ void k_view_knn(const float* views, const float* vt, int* view_inds, int B) {
    int id = blockIdx.x * blockDim.x + threadIdx.x;
    if (id >= B * NV) return;
    int b = id / NV, v = id % NV;
    float px = views[v*3], py = views[v*3+1], pz = views[v*3+2];
    const float* q = vt + (size_t)b * NV * 3;
    float best = 3.0e38f; int bi = 0;
    for (int u = 0; u < NV; ++u) {
        float dx = px - q[u*3], dy = py - q[u*3+1], dz = pz - q[u*3+2];
        float d = dx*dx + dy*dy + dz*dz;
        if (d < best) { best = d; bi = u; }
    }
    view_inds[id] = bi;
}

// ---------------------------------------------------------------- rot_trans = R @ GR[view_inds]
__global__ void k_rot_trans(const float* pose, const float* GRm, const int* view_inds,
                            float* rot_trans, int B) {
    int id = blockIdx.x * blockDim.x + threadIdx.x;
    if (id >= B * VAE) return;
    int b = id / VAE, va = id % VAE;
    int v = va / NA, a = va % NA;
    const float* P = pose + (size_t)b * 12;
    int g = view_inds[b*NV + v];
    const float* Gm = GRm + ((size_t)g * NA + a) * 9;
    float* Rt = rot_trans + (size_t)id * 9;
    #pragma unroll
    for (int i = 0; i < 3; ++i)
        #pragma unroll
        for (int j = 0; j < 3; ++j)
            Rt[i*3+j] = P[i*4+0]*Gm[0*3+j] + P[i*4+1]*Gm[1*3+j] + P[i*4+2]*Gm[2*3+j];
}

// ---------------------------------------------------------------- angle alignment
__global__ void k_align(const float* pkbuf, const float* rot_trans,
                        int* angle_inds, float* rot_out, int B) {
    int id = blockIdx.x * blockDim.x + threadIdx.x;
    if (id >= B * VAE) return;
    int b = id / VAE, va = id % VAE;
    int v = va / NA, a = va % NA;
    const float* pk = pkbuf + ((size_t)v * NA + a) * 12;
    const float lx[4] = {0.02f, 0.02f, 0.02f, 0.0f};
    const float ly[4] = {0.0f,  0.01f, -0.01f, 0.0f};
    float best1 = 3.0e38f, best2 = 3.0e38f; int i1 = 0, i2 = 0;
    for (int a2 = 0; a2 < NA; ++a2) {
        const float* Rm = rot_trans + (((size_t)b * VAE) + v*NA + a2) * 9;
        float d1 = 0.0f, d2 = 0.0f;
        #pragma unroll
        for (int k = 0; k < 4; ++k) {
            #pragma unroll
            for (int i = 0; i < 3; ++i) {
                float t  = Rm[i*3+0]*lx[k] + Rm[i*3+1]*ly[k];
                float ts = Rm[i*3+0]*lx[k] - Rm[i*3+1]*ly[k];   // sym: ly -> -ly
                float ref = pk[k*3 + i];
                float e1 = ref - t;  d1 += e1*e1;
                float e2 = ref - ts; d2 += e2*e2;
            }
        }
        if (d1 < best1) { best1 = d1; i1 = a2; }
        if (d2 < best2) { best2 = d2; i2 = a2; }
    }
    int ind = (best1 < best2) ? i1 : i2;
    angle_inds[id] = ind;
    const float* src = rot_trans + (((size_t)b * VAE) + v*NA + ind) * 9;
    float* dst = rot_out + (size_t)id * 9;
    #pragma unroll
    for (int k = 0; k < 9; ++k) dst[k] = src[k];
}

// ---------------------------------------------------------------- gpoints_trans = gp @ R^T + t
__global__ void k_gpoints_trans(const float* pose, const float* gp, float* gpt, int B) {
    int id = blockIdx.x * blockDim.x + threadIdx.x;
    if (id >= B * NP) return;
    int b = id / NP;
    const float* P = pose + (size_t)b * 12;
    const float* g = gp + (size_t)id * 3;
    #pragma unroll
    for (int i = 0; i < 3; ++i)
        gpt[(size_t)id*3 + i] = g[0]*P[i*4+0] + g[1]*P[i*4+1] + g[2]*P[i*4+2] + P[i*4+3];
}

// ---------------------------------------------------------------- padded (x,y,z,|p|^2) table
__global__ void k_pad_gpt(const float* gpt, float4* gp4, int B) {
    int id = blockIdx.x * blockDim.x + threadIdx.x;
    if (id >= B * NPPAD) return;
    int b = id / NPPAD, p = id % NPPAD;
    float4 o;
    if (p < NP) {
        const float* g = gpt + ((size_t)b * NP + p) * 3;
        o.x = g[0]; o.y = g[1]; o.z = g[2];
        o.w = o.x*o.x + o.y*o.y + o.z*o.z;
    } else {
        o.x = 0.0f; o.y = 0.0f; o.z = 0.0f; o.w = 1.0e30f;   // never selected
    }
    gp4[id] = o;
}

// ---------------------------------------------------------------- WMMA seed->gpoint NN
// Distance dot-products via V_WMMA_F32_16X16X4_F32 (16 seeds x 16 points, K=3 pad 4).
// argmin over p is invariant to |s|^2, so each lane tracks |p|^2 - 2*dot.
// The padded (x,y,z,|p|^2) table is staged in LDS once per block; the inner loop is
// branchless ds_load_b128 + 4x back-to-back WMMA (full EXEC as WMMA requires).
__global__ void k_nn_wmma(const float* seeds, const float4* gp4, int* nn_inds, int B) {
    __shared__ float4 sg[NPPAD];                   // 24 KB of 320 KB WGP LDS
    int b = blockIdx.x >> 3;                       // 8 blocks per batch
    for (int i = threadIdx.x; i < NPPAD; i += 256)
        sg[i] = gp4[(size_t)b * NPPAD + i];
    __syncthreads();

    int lane = threadIdx.x & 31;
    int m0 = (((blockIdx.x & 7) << 3) + (threadIdx.x >> 5)) << 4;  // seed tile * 16
    bool lo = lane < 16;
    int col = lane & 15;
    // A (16x4): lane L holds row M=L%16; VGPR0: K = lo?0:2, VGPR1: K = lo?1:3 (K=3 -> 0)
    const float* sp = seeds + ((size_t)b * NS + m0 + col) * 3;
    float sx = sp[0], sy = sp[1], sz = sp[2];
    v2f A; A.x = lo ? sx : sz; A.y = lo ? sy : 0.0f;

    float best[8]; int bidx[8];
    #pragma unroll
    for (int j = 0; j < 8; ++j) { best[j] = 3.0e38f; bidx[j] = 0; }

    for (int t = 0; t < NTILE; t += 4) {
        #pragma unroll
        for (int u = 0; u < 4; ++u) {
            int p = (t + u) * 16 + col;
            float4 g = sg[p];
            // B (4x16): lane L holds col N=L%16; VGPR0: K = lo?0:2, VGPR1: K = lo?1:3
            v2f Bv; Bv.x = lo ? g.x : g.z; Bv.y = lo ? g.y : 0.0f;
            v8f C = {};
            C = __builtin_amdgcn_wmma_f32_16x16x4_f32(false, A, false, Bv, (short)0, C,
                                                      false, false);
            float pn = g.w;
            #pragma unroll
            for (int j = 0; j < 8; ++j) {
                float d = pn - 2.0f * C[j];
                if (d < best[j]) { best[j] = d; bidx[j] = p; }   // ascending p => first-min
            }
        }
    }
    // reduce min/argmin across the 16 lanes of each half-wave (C layout: lane = N col)
    #pragma unroll
    for (int off = 1; off < 16; off <<= 1) {
        #pragma unroll
        for (int j = 0; j < 8; ++j) {
            float ov = __shfl_xor(best[j], off, 32);
            int   oi = __shfl_xor(bidx[j], off, 32);
            if (ov < best[j] || (ov == best[j] && oi < bidx[j])) { best[j] = ov; bidx[j] = oi; }
        }
    }
    if (col == 0) {
        int rbase = m0 + (lo ? 0 : 8);   // VGPR j: M = j (lanes 0-15) / j+8 (lanes 16-31)
        #pragma unroll
        for (int j = 0; j < 8; ++j) nn_inds[b*NS + rbase + j] = bidx[j];
    }
}

// ---------------------------------------------------------------- fused gather + graspness + tops
__global__ void k_fused(const float* gscores, const float* gwidths,
                        const int* nn_inds, const int* view_inds, const int* angle_inds,
                        const float* rot_out, const float* gpt, const int* top_inds,
                        float* out, int B) {
    __shared__ float raw[VAE];
    __shared__ float red_mn[256];
    __shared__ float red_mx[256];
    int bs = blockIdx.x;
    int b = bs / NS, s = bs % NS;
    int tid = threadIdx.x;
    int p = nn_inds[b*NS + s];
    const float* sc_base = gscores + (((size_t)b * NP + p) * NV) * (NA * ND);
    float mn = 3.0e38f, mx = -3.0e38f;
    for (int va = tid; va < VAE; va += blockDim.x) {
        int v = va / NA;
        int sv = view_inds[b*NV + v];
        int sa = angle_inds[(size_t)b*VAE + va];
        const float4 c4 = *(const float4*)(sc_base + ((size_t)sv * NA + sa) * ND);
        float cnt = ((c4.x > 0.0f && c4.x <= 0.6f) ? 1.0f : 0.0f)
                  + ((c4.y > 0.0f && c4.y <= 0.6f) ? 1.0f : 0.0f)
                  + ((c4.z > 0.0f && c4.z <= 0.6f) ? 1.0f : 0.0f)
                  + ((c4.w > 0.0f && c4.w <= 0.6f) ? 1.0f : 0.0f);
        float g = cnt * 0.25f;
        raw[va] = g;
        mn = fminf(mn, g); mx = fmaxf(mx, g);
    }
    red_mn[tid] = mn; red_mx[tid] = mx;
    __syncthreads();
    for (int st = 128; st > 0; st >>= 1) {
        if (tid < st) {
            red_mn[tid] = fminf(red_mn[tid], red_mn[tid + st]);
            red_mx[tid] = fmaxf(red_mx[tid], red_mx[tid + st]);
        }
        __syncthreads();
    }
    float gmn = red_mn[0], gmx = red_mx[0];
    float den = gmx - gmn + 1e-8f;
    size_t ob = ((size_t)b * NS + s) * OUTW;
    for (int va = tid; va < VAE; va += blockDim.x)
        out[ob + 8 + va] = (raw[va] - gmn) / den;
    if (tid == 0) {
        int top = top_inds[b*NS + s];
        int v = top / NA;
        int sv = view_inds[b*NV + v];
        int sa = angle_inds[(size_t)b*VAE + top];
        const float* sp = sc_base + ((size_t)sv * NA + sa) * ND;
        const float* wp = gwidths + ((((size_t)b * NP + p) * NV + sv) * NA + sa) * ND;
        #pragma unroll
        for (int d = 0; d < ND; ++d) {
            float sc = sp[d], wd = wp[d];
            out[ob + 4 + d] = wd;
            out[ob + d] = (sc > 0.0f && wd <= 0.1f) ? sc : 0.0f;   // lmask
        }
    }
    if (tid == 1) {
        int top = top_inds[b*NS + s];
        const float* r = rot_out + ((size_t)b * VAE + top) * 9;
        #pragma unroll
        for (int k = 0; k < 9; ++k) out[ob + 3608 + k] = r[k];
    }
    if (tid == 2) {
        const float* g = gpt + ((size_t)b * NP + p) * 3;
        #pragma unroll
        for (int k = 0; k < 3; ++k) out[ob + 3617 + k] = g[k];
    }
}

// ---------------------------------------------------------------- global u_max / u_min(+)
__global__ void k_ustats(const float* out, float* ustats, int B) {
    __shared__ float smx[256];
    __shared__ float smn[256];
    int tid = threadIdx.x;
    int total = B * NS * ND;
    float mx = -3.0e38f, mnp = 3.0e38f;
    for (int i = tid; i < total; i += 256) {
        int row = i >> 2, d = i & 3;
        float x = out[(size_t)row * OUTW + d];
        mx = fmaxf(mx, x);
        if (x > 0.0f) mnp = fminf(mnp, x);
    }
    smx[tid] = mx; smn[tid] = mnp;
    __syncthreads();
    for (int st = 128; st > 0; st >>= 1) {
        if (tid < st) {
            smx[tid] = fmaxf(smx[tid], smx[tid + st]);
            smn[tid] = fminf(smn[tid], smn[tid + st]);
        }
        __syncthreads();
    }
    if (tid == 0) {
        ustats[0] = smx[0];
        ustats[1] = (smn[0] > 2.0e38f) ? 1.0f : smn[0];   // no positives -> u_min = 1.0
    }
}

__global__ void k_norm(float* out, const float* ustats, int B) {
    int i = blockIdx.x * blockDim.x + threadIdx.x;
    if (i >= B * NS * ND) return;
    int row = i >> 2, d = i & 3;
    size_t idx = (size_t)row * OUTW + d;
    float x = out[idx];
    if (x > 0.0f) {
        float umax = ustats[0], umin = ustats[1];
        out[idx] = logf(umax / x) / (logf(umax / umin) + 1e-8f);
    }
}

// ---------------------------------------------------------------- host
extern "C" void kernel_launch(void* const* d_in, const int* in_sizes, int n_in,
                              void* d_out, int out_size, void* d_ws, size_t ws_size,
                              hipStream_t stream) {
    const float* point_clouds = (const float*)d_in[0];
    const float* object_pose  = (const float*)d_in[1];
    const float* grasp_points = (const float*)d_in[2];
    const float* grasp_labels = (const float*)d_in[3];
    const float* grasp_offs   = (const float*)d_in[4];
    const int*   top_inds     = (const int*)d_in[5];
    float* out = (float*)d_out;
    int B = in_sizes[0] / (NS * 3);

    // workspace layout (4-byte units; gp4 first for 16B alignment)
    float*  ws         = (float*)d_ws;
    float4* gp4        = (float4*)ws;                        // B*1536 float4
    float*  views      = ws + (size_t)B * NPPAD * 4;         // 300*3
    float*  GRm        = views + NV*3;                       // 3600*9
    float*  pkbuf      = GRm + (size_t)VAE*9;                // 3600*12
    float*  vt         = pkbuf + (size_t)VAE*12;             // B*300*3
    int*    view_inds  = (int*)(vt + (size_t)B*NV*3);        // B*300
    float*  rot_trans  = (float*)(view_inds + (size_t)B*NV); // B*3600*9
    int*    angle_inds = (int*)(rot_trans + (size_t)B*VAE*9);// B*3600
    float*  rot_out    = (float*)(angle_inds + (size_t)B*VAE);// B*3600*9
    float*  gpt        = rot_out + (size_t)B*VAE*9;          // B*1500*3
    int*    nn_inds    = (int*)(gpt + (size_t)B*NP*3);       // B*1024
    float*  ustats     = (float*)(nn_inds + (size_t)B*NS);   // 2

    const int T = 256;
    k_setup<<<(VAE + T - 1)/T, T, 0, stream>>>(views, GRm, pkbuf);
    k_views_trans<<<(B*NV + T - 1)/T, T, 0, stream>>>(object_pose, views, vt, B);
    k_view_knn<<<(B*NV + T - 1)/T, T, 0, stream>>>(views, vt, view_inds, B);
    k_rot_trans<<<(B*VAE + T - 1)/T, T, 0, stream>>>(object_pose, GRm, view_inds, rot_trans, B);
    k_align<<<(B*VAE + T - 1)/T, T, 0, stream>>>(pkbuf, rot_trans, angle_inds, rot_out, B);
    k_gpoints_trans<<<(B*NP + T - 1)/T, T, 0, stream>>>(object_pose, grasp_points, gpt, B);
    k_pad_gpt<<<(B*NPPAD + T - 1)/T, T, 0, stream>>>(gpt, gp4, B);
    // B*64 seed tiles, 8 waves (tiles) per 256-thread block
    k_nn_wmma<<<B*8, T, 0, stream>>>(point_clouds, gp4, nn_inds, B);
    k_fused<<<B*NS, T, 0, stream>>>(grasp_labels, grasp_offs, nn_inds, view_inds,
                                    angle_inds, rot_out, gpt, top_inds, out, B);
    k_ustats<<<1, T, 0, stream>>>(out, ustats, B);
    k_norm<<<(B*NS*ND + T - 1)/T, T, 0, stream>>>(out, ustats, B);
}